// FeatureDecorr_v3_49271864820158
// MI455X (gfx1250) — compile-verified
//
#include <hip/hip_runtime.h>
#include <hip/hip_bf16.h>
#include <cstddef>

// ---------------------------------------------------------------------------
// FeatureDecorr (group whitening) for MI455X / gfx1250.
//
// x: (32, 256, 56, 56) fp32. Group g = channel % 16. M = 1,605,632 per group.
//
// Pass 1: Gram S = x·xT via V_WMMA_F32_16X16X4_F32, feeding the SAME regs as
//         A and B (A layout: row=lane&15, K striped over {vgpr,half}; B is the
//         mirror with col=lane&15) -> D += tile @ tileT. Per-lane running
//         sums give means. Each wave owns 98 contiguous 4-position tiles
//         inside ONE slab (784/98 = 8 waves/slab) so all address math hoists
//         and loads use immediate offsets. Deterministic LDS block reduction.
// Pass 2: one block: sequential reduction of 512 partials, cov = S/M - m·mT
//         + eps·I, 10 Newton-Schulz iterations in ONE wave (12 fp32 WMMAs
//         per iteration, operands staged via LDS). Emits decorr + mean.
// Pass 3: y = decorr @ (x - mean) per 16x16 tile, 4 chained WMMAs, fused
//         weight/bias. x fits in 192MB L2 after pass 1; pass-3 reads use NT
//         (last use) and y stores use NT so the streaming output does NOT
//         evict x -> ~210MB HBM traffic instead of ~310MB (~9us floor at
//         23.3 TB/s). 49 waves/slab * 4 tiles keeps slab math loop-invariant.
// ---------------------------------------------------------------------------

typedef __attribute__((ext_vector_type(2))) float v2f;
typedef __attribute__((ext_vector_type(8))) float v8f;

#define HW 3136            // 56*56
#define CHW 802816         // 256*3136
#define GRP 16
#define M_TOT 1605632.0f   // 32*16*3136
#define EPSV 1e-5f

#define P1_BLOCKS 512
#define P1_TPW    98       // tiles(4 pos) per wave; 4096 waves * 98 = 401408
#define PART_STRIDE 288    // 256 cov + 32 lane-sums per block partial

#define P3_BLOCKS 3136     // 25088 waves * 4 tiles(16 pos) = 100352
#define P3_TPW    4

__device__ __forceinline__ v8f wmma4(v2f a, v2f b, v8f c) {
  // D(16x16,f32) += A(16x4,f32) * B(4x16,f32)
  return __builtin_amdgcn_wmma_f32_16x16x4_f32(
      /*neg_a=*/false, a, /*neg_b=*/false, b,
      /*c_mod=*/(short)0, c, /*reuse_a=*/false, /*reuse_b=*/false);
}

// ---------------------------------------------------------------- pass 1 ----
__global__ void __launch_bounds__(256)
fd_pass1_gram(const float* __restrict__ x, float* __restrict__ part) {
  __shared__ float red[8][32][9];
  const int tid  = threadIdx.x;
  const int wave = tid >> 5;
  const int lane = tid & 31;
  const int r    = lane & 15;      // group row (A: M index, B: N index)
  const int hi   = lane >> 4;      // K half select
  const int wg   = blockIdx.x * 8 + wave;   // 0..4095

  // 8 waves per slab, each owning 98*4 = 392 contiguous positions.
  const int slab = wg >> 3;                 // (n, c_outer), loop-invariant
  const int pst  = (wg & 7) * (P1_TPW * 4); // start position in slab
  const float* px = x + (size_t)(slab >> 4) * CHW +
                    (size_t)((slab & 15) * GRP + r) * HW + pst + 2 * hi;

  v8f acc0 = {}; v8f acc1 = {};
  float lsum0 = 0.f, lsum1 = 0.f;

  #pragma unroll 4
  for (int it = 0; it < P1_TPW; it += 2) {
    __builtin_prefetch(px + it * 4 + 512, 0, 3);   // global_prefetch_b8
    v2f a0 = *(const v2f*)(px + it * 4);
    v2f a1 = *(const v2f*)(px + it * 4 + 4);
    lsum0 += a0.x + a0.y;
    acc0 = wmma4(a0, a0, acc0);                    // += tile @ tileT
    lsum1 += a1.x + a1.y;
    acc1 = wmma4(a1, a1, acc1);
  }
  acc0 += acc1;
  lsum0 += lsum1;

  #pragma unroll
  for (int v = 0; v < 8; ++v) red[wave][lane][v] = acc0[v];
  red[wave][lane][8] = lsum0;
  __syncthreads();

  // deterministic 8-wave reduction -> one block partial
  {
    const int v = tid >> 5, ln = tid & 31;
    float s = 0.f;
    #pragma unroll
    for (int w = 0; w < 8; ++w) s += red[w][ln][v];
    part[(size_t)blockIdx.x * PART_STRIDE + v * 32 + ln] = s;
  }
  if (tid < 32) {
    float s = 0.f;
    #pragma unroll
    for (int w = 0; w < 8; ++w) s += red[w][tid][8];
    part[(size_t)blockIdx.x * PART_STRIDE + 256 + tid] = s;
  }
}

// ---------------------------------------------------------------- pass 2 ----
__global__ void __launch_bounds__(256)
fd_pass2_ns(const float* __restrict__ part, float* __restrict__ dm) {
  __shared__ float Smat[16][16];
  __shared__ float Amat[16][16];
  __shared__ float Ymat[16][17];
  __shared__ float Zmat[16][17];
  __shared__ float Tmat[16][17];
  __shared__ float meanL[16];
  __shared__ float red[256];
  __shared__ float normL;

  const int tid = threadIdx.x;

  // sequential, deterministic reduction of block partials
  {
    float s = 0.f;
    for (int b = 0; b < P1_BLOCKS; ++b) s += part[(size_t)b * PART_STRIDE + tid];
    const int v = tid >> 5, ln = tid & 31;
    // C/D layout: element (M = v + 8*(ln>>4), N = ln&15)
    Smat[v + 8 * (ln >> 4)][ln & 15] = s;
  }
  if (tid < 16) {
    float s = 0.f;
    for (int b = 0; b < P1_BLOCKS; ++b) {
      const float* p = part + (size_t)b * PART_STRIDE + 256;
      s += p[tid] + p[16 + tid];   // K={0,1} half + K={2,3} half
    }
    meanL[tid] = s * (1.0f / M_TOT);
  }
  __syncthreads();

  const int i = tid >> 4, j = tid & 15;
  const float a = Smat[i][j] * (1.0f / M_TOT) - meanL[i] * meanL[j] +
                  (i == j ? EPSV : 0.f);
  Amat[i][j] = a;
  red[tid] = a * a;
  __syncthreads();
  if (tid == 0) {
    float s = 0.f;
    for (int k = 0; k < 256; ++k) s += red[k];
    normL = sqrtf(s);              // Frobenius norm
  }
  __syncthreads();

  Ymat[i][j] = Amat[i][j] / normL;
  Zmat[i][j] = (i == j) ? 1.f : 0.f;
  __syncthreads();

  // 10 Newton-Schulz iterations, single wave, all matmuls on WMMA fp32.
  if (tid < 32) {
    const int r  = tid & 15;   // A row / B,D column
    const int hi = tid >> 4;
    for (int it = 0; it < 10; ++it) {
      // ZY = Z @ Y
      v8f zy = {};
      #pragma unroll
      for (int k0 = 0; k0 < 16; k0 += 4) {
        const int k = k0 + 2 * hi;
        v2f av; av.x = Zmat[r][k]; av.y = Zmat[r][k + 1];
        v2f bv; bv.x = Ymat[k][r]; bv.y = Ymat[k + 1][r];
        zy = wmma4(av, bv, zy);
      }
      // T = 0.5*(3I - ZY)
      #pragma unroll
      for (int v = 0; v < 8; ++v) {
        const int mrow = v + 8 * hi;
        Tmat[mrow][r] = 0.5f * ((mrow == r ? 3.f : 0.f) - zy[v]);
      }
      // Ynew = Y @ T ; Znew = T @ Z   (loads of old Y/Z precede stores: OK,
      // LDS is in-order within a wave)
      v8f ay = {}, az = {};
      #pragma unroll
      for (int k0 = 0; k0 < 16; k0 += 4) {
        const int k = k0 + 2 * hi;
        v2f a1; a1.x = Ymat[r][k]; a1.y = Ymat[r][k + 1];
        v2f b1; b1.x = Tmat[k][r]; b1.y = Tmat[k + 1][r];
        ay = wmma4(a1, b1, ay);
        v2f a2; a2.x = Tmat[r][k]; a2.y = Tmat[r][k + 1];
        v2f b2; b2.x = Zmat[k][r]; b2.y = Zmat[k + 1][r];
        az = wmma4(a2, b2, az);
      }
      #pragma unroll
      for (int v = 0; v < 8; ++v) {
        const int mrow = v + 8 * hi;
        Ymat[mrow][r] = ay[v];
        Zmat[mrow][r] = az[v];
      }
    }
  }
  __syncthreads();

  const float sc = 1.0f / sqrtf(normL);
  dm[tid] = Zmat[i][j] * sc;            // decorr row-major [out_g][in_g]
  if (tid < 16) dm[256 + tid] = meanL[tid];
}

// ---------------------------------------------------------------- pass 3 ----
__global__ void __launch_bounds__(256)
fd_pass3_whiten(const float* __restrict__ x, const float* __restrict__ wgt,
                const float* __restrict__ bia, const float* __restrict__ dm,
                float* __restrict__ out) {
  __shared__ float dec[16][16];
  __shared__ float mn[16];
  __shared__ float wl[256];
  __shared__ float bl[256];
  const int tid = threadIdx.x;
  dec[tid >> 4][tid & 15] = dm[tid];
  wl[tid] = wgt[tid];
  bl[tid] = bia[tid];
  if (tid < 16) mn[tid] = dm[256 + tid];
  __syncthreads();

  const int wave = tid >> 5;
  const int lane = tid & 31;
  const int c    = lane & 15;    // A row (= out-group stripe) & B/D column
  const int hi   = lane >> 4;
  const int wg   = blockIdx.x * 8 + wave;    // 0..25087

  // 49 waves per slab * 4 tiles = 196 tiles: slab math is loop-invariant.
  const int slab = wg / 49;
  const int tloc = wg % 49;
  const int cg   = (slab & 15) * GRP;        // channel-group base
  const size_t sbase = (size_t)(slab >> 4) * CHW + (size_t)cg * HW;
  const float* px = x + sbase + tloc * 64 + c;
  float* py      = out + sbase + tloc * 64 + c;

  for (int it = 0; it < P3_TPW; ++it) {
    const int p0 = it * 16;
    v8f acc = {};
    #pragma unroll
    for (int k0 = 0; k0 < 16; k0 += 4) {
      const int k = k0 + 2 * hi;
      v2f av; av.x = dec[c][k]; av.y = dec[c][k + 1];        // A = decorr
      v2f bv;                                                // B = xc tile
      // last-use reads: x was brought into L2 by pass 1; NT avoids re-pin
      bv.x = __builtin_nontemporal_load(px + (size_t)k * HW + p0)       - mn[k];
      bv.y = __builtin_nontemporal_load(px + (size_t)(k + 1) * HW + p0) - mn[k + 1];
      acc = wmma4(av, bv, acc);
    }
    #pragma unroll
    for (int v = 0; v < 8; ++v) {
      const int mrow = v + 8 * hi;             // output group
      const int ch   = cg + mrow;              // channel
      // NT store: y is write-once streaming; don't evict x from L2
      __builtin_nontemporal_store(acc[v] * wl[ch] + bl[ch],
                                  py + (size_t)mrow * HW + p0);
    }
  }
}

// --------------------------------------------------------------- launch ----
extern "C" void kernel_launch(void* const* d_in, const int* in_sizes, int n_in,
                              void* d_out, int out_size, void* d_ws, size_t ws_size,
                              hipStream_t stream) {
  const float* x   = (const float*)d_in[0];
  const float* wgt = (const float*)d_in[1];   // (1,256,1,1) flat
  const float* bia = (const float*)d_in[2];
  float* out = (float*)d_out;

  float* part = (float*)d_ws;                              // 512*288 floats
  float* dm   = part + (size_t)P1_BLOCKS * PART_STRIDE;    // 256 decorr + 16 mean

  fd_pass1_gram<<<P1_BLOCKS, 256, 0, stream>>>(x, part);
  fd_pass2_ns<<<1, 256, 0, stream>>>(part, dm);
  fd_pass3_whiten<<<P3_BLOCKS, 256, 0, stream>>>(x, wgt, bia, dm, out);
}